// DNC_73057393705413
// MI455X (gfx1250) — compile-verified
//
#include <hip/hip_runtime.h>
#include <math.h>

// ---------------- problem constants ----------------
#define TST   64
#define BB    16
#define INS   256
#define OUTS  256
#define WLEN  128
#define NC    256
#define RR    4
#define HIDN  512
#define CTRLR 927          // real control width
#define CTRLP 928          // padded (58 tiles of 16)
#define KG    1280         // gates GEMM K = 768 (ctrl_in) + 512 (h)
#define NG    2048         // 4*HID
#define KO    1024         // out GEMM K = 512 (h) + 512 (r_data)
#define CLIPV 20.0f
#define EPSF  1e-6f

typedef float v2f_t __attribute__((ext_vector_type(2)));
typedef float v8f_t __attribute__((ext_vector_type(8)));

// ---------------- device helpers ----------------
__device__ __forceinline__ float sigm(float x) { return 1.0f / (1.0f + expf(-x)); }
// inputs are pre-clipped to [-20,20], so plain log1p(exp(x)) is safe
__device__ __forceinline__ float oneplus_f(float x) { return log1pf(expf(x)) + 1.0f; }

// block == 256 threads
__device__ __forceinline__ float bmax256(float v, float* s) {
  int t = threadIdx.x;
  s[t] = v; __syncthreads();
  for (int off = 128; off > 0; off >>= 1) {
    if (t < off) s[t] = fmaxf(s[t], s[t + off]);
    __syncthreads();
  }
  float r = s[0]; __syncthreads();
  return r;
}
__device__ __forceinline__ float bsum256(float v, float* s) {
  int t = threadIdx.x;
  s[t] = v; __syncthreads();
  for (int off = 128; off > 0; off >>= 1) {
    if (t < off) s[t] += s[t + off];
    __syncthreads();
  }
  float r = s[0]; __syncthreads();
  return r;
}

// ---------------- WMMA f32 GEMM: C[16,N] = A[16,K] @ W[K,N] (+bias, optional clip)
// one wave per 16x16 C tile; K stepped by 4 with V_WMMA_F32_16X16X4_F32.
// A frag per ISA layout: lane = m + 16*(k>=2), vgpr = k&1.
// B frag analogue:       lane = n + 16*(k>=2), vgpr = k&1.
__global__ void dnc_gemm(const float* __restrict__ A, const float* __restrict__ W,
                         const float* __restrict__ bias, float* __restrict__ C,
                         int K, int N, int doclip)
{
  int lane   = threadIdx.x & 31;
  int wave   = threadIdx.x >> 5;
  int ntiles = (N + 15) >> 4;
  int tile   = blockIdx.x * (blockDim.x >> 5) + wave;
  if (tile >= ntiles) return;          // wave-uniform; EXEC stays all-ones inside
  int half = lane >> 4;                // 0: k-offset 0, 1: k-offset 2
  int lm   = lane & 15;                // m for A-frag, n for B-frag
  int n0   = tile << 4;

  const float* arow = A + (size_t)lm * K + 2 * half;
  const float* wcol = W + (size_t)(2 * half) * N + n0 + lm;
  __builtin_prefetch(arow, 0, 1);      // global_prefetch_b8
  __builtin_prefetch(wcol, 0, 1);

  v8f_t acc = {};
  for (int k = 0; k < K; k += 4) {
    v2f_t af, bf;
    af.x = arow[k];
    af.y = arow[k + 1];
    bf.x = wcol[(size_t)k * N];
    bf.y = wcol[(size_t)(k + 1) * N];
    acc = __builtin_amdgcn_wmma_f32_16x16x4_f32(
        false, af, false, bf, (short)0, acc, false, false);
  }
  float bv = bias ? bias[n0 + lm] : 0.0f;
  for (int r = 0; r < 8; ++r) {
    int m = r + 8 * half;              // C layout: vgpr r, lanes>=16 => M += 8
    float v = acc[r] + bv;
    if (doclip) v = fminf(fmaxf(v, -CLIPV), CLIPV);
    C[(size_t)m * N + n0 + lm] = v;
  }
}

// WMMA fwd/bwd einsum: fwd[b,h,i] = sum_j rd[b,h,j] * link[b,i,j]
//                      bwd[b,h,i] = sum_j rd[b,h,j] * link[b,j,i]
// M padded 4->16 (A rows >=4 fed zeros via lane mask; padded rows discarded at store).
// 512 waves total: wave id = b*32 + dir*16 + tile(of NC).
__global__ void dnc_fb_gemm(const float* __restrict__ rd, const float* __restrict__ link,
                            float* __restrict__ fwd, float* __restrict__ bwd)
{
  int lane = threadIdx.x & 31;
  int wave = threadIdx.x >> 5;
  int wid  = blockIdx.x * (blockDim.x >> 5) + wave;
  int b    = wid >> 5;
  int rem  = wid & 31;
  int dir  = rem >> 4;                 // 0 = fwd, 1 = bwd
  int tile = rem & 15;
  int half = lane >> 4;
  int lm   = lane & 15;
  int n0   = tile << 4;

  int   mrow  = (lm < RR) ? lm : (RR - 1);          // clamp: no OOB loads
  float amask = (lm < RR) ? 1.0f : 0.0f;            // zero padded A rows
  const float* arow = rd + ((size_t)b * RR + mrow) * NC + 2 * half;
  const float* Lb   = link + (size_t)b * NC * NC;

  v8f_t acc = {};
  if (dir == 0) {                      // fwd: B(k,n) = link[n,k] (contiguous pairs)
    const float* brow = Lb + (size_t)(n0 + lm) * NC + 2 * half;
    for (int k = 0; k < NC; k += 4) {
      v2f_t af, bf;
      af.x = amask * arow[k];
      af.y = amask * arow[k + 1];
      bf.x = brow[k];
      bf.y = brow[k + 1];
      acc = __builtin_amdgcn_wmma_f32_16x16x4_f32(
          false, af, false, bf, (short)0, acc, false, false);
    }
  } else {                             // bwd: B(k,n) = link[k,n] (column walk)
    const float* bcol = Lb + (size_t)(2 * half) * NC + n0 + lm;
    for (int k = 0; k < NC; k += 4) {
      v2f_t af, bf;
      af.x = amask * arow[k];
      af.y = amask * arow[k + 1];
      bf.x = bcol[(size_t)k * NC];
      bf.y = bcol[(size_t)(k + 1) * NC];
      acc = __builtin_amdgcn_wmma_f32_16x16x4_f32(
          false, af, false, bf, (short)0, acc, false, false);
    }
  }
  float* out = (dir == 0) ? fwd : bwd;
  if (half == 0) {                     // rows m=0..3 live in vgprs 0..3, lanes 0..15
    for (int r = 0; r < RR; ++r)
      out[((size_t)b * RR + r) * NC + n0 + lm] = acc[r];
  }
}

// WMMA read-data einsum: r_data[b,h,w] = sum_n rd[b,h,n] * mem[b,n,w]
// 128 waves: wave id = b*8 + tile(of WLEN). Also packs A_out[512:1024].
__global__ void dnc_rdata_gemm(const float* __restrict__ rd, const float* __restrict__ mem,
                               float* __restrict__ rdata, float* __restrict__ Aout)
{
  int lane = threadIdx.x & 31;
  int wave = threadIdx.x >> 5;
  int wid  = blockIdx.x * (blockDim.x >> 5) + wave;
  int b    = wid >> 3;
  int tile = wid & 7;
  int half = lane >> 4;
  int lm   = lane & 15;
  int w0   = tile << 4;

  int   mrow  = (lm < RR) ? lm : (RR - 1);
  float amask = (lm < RR) ? 1.0f : 0.0f;
  const float* arow = rd + ((size_t)b * RR + mrow) * NC + 2 * half;
  const float* bcol = mem + ((size_t)b * NC + 2 * half) * WLEN + w0 + lm;

  v8f_t acc = {};
  for (int k = 0; k < NC; k += 4) {
    v2f_t af, bf;
    af.x = amask * arow[k];
    af.y = amask * arow[k + 1];
    bf.x = bcol[(size_t)k * WLEN];
    bf.y = bcol[(size_t)(k + 1) * WLEN];
    acc = __builtin_amdgcn_wmma_f32_16x16x4_f32(
        false, af, false, bf, (short)0, acc, false, false);
  }
  if (half == 0) {
    for (int r = 0; r < RR; ++r) {
      float v = acc[r];
      rdata[(size_t)b * (RR * WLEN) + r * WLEN + w0 + lm] = v;      // prev_rdata
      Aout[(size_t)b * KO + HIDN + r * WLEN + w0 + lm]    = v;
    }
  }
}

// ---------------- pre-pack / init kernels ----------------
__global__ void dnc_zero(float* p, int n) {
  int i = blockIdx.x * 256 + threadIdx.x;
  if (i < n) p[i] = 0.0f;
}
__global__ void dnc_pack_w1(const float* __restrict__ Wx, const float* __restrict__ Wh,
                            float* __restrict__ W1) {
  size_t i = (size_t)blockIdx.x * 256 + threadIdx.x;
  if (i >= (size_t)KG * NG) return;
  size_t r = i / NG, c = i % NG;
  W1[i] = (r < 768) ? Wx[r * NG + c] : Wh[(r - 768) * NG + c];
}
__global__ void dnc_pack_w2(const float* __restrict__ Wc, float* __restrict__ W2) {
  size_t i = (size_t)blockIdx.x * 256 + threadIdx.x;
  if (i >= (size_t)HIDN * CTRLP) return;
  size_t r = i / CTRLP, c = i % CTRLP;
  W2[i] = (c < CTRLR) ? Wc[r * CTRLR + c] : 0.0f;
}
__global__ void dnc_pack_bc(const float* __restrict__ bc, float* __restrict__ bcp) {
  int i = blockIdx.x * 256 + threadIdx.x;
  if (i < CTRLP) bcp[i] = (i < CTRLR) ? bc[i] : 0.0f;
}
__global__ void dnc_pack_w3(const float* __restrict__ Wo, const float* __restrict__ Wr,
                            float* __restrict__ W3) {
  size_t i = (size_t)blockIdx.x * 256 + threadIdx.x;
  if (i >= (size_t)KO * OUTS) return;
  size_t r = i / OUTS, c = i % OUTS;
  W3[i] = (r < 512) ? Wo[r * OUTS + c] : Wr[(r - 512) * OUTS + c];
}
__global__ void dnc_pack_bo(const float* __restrict__ bo, const float* __restrict__ br,
                            float* __restrict__ bop) {
  int i = threadIdx.x;
  if (i < OUTS) bop[i] = bo[i] + br[i];
}

// ---------------- per-step kernels ----------------
// A_gates[b, 0:256]=x_t, [256:768]=prev_rdata, [768:1280]=h
__global__ void dnc_packA(const float* __restrict__ xt, const float* __restrict__ rdata,
                          const float* __restrict__ h, float* __restrict__ Ag) {
  int i = blockIdx.x * 256 + threadIdx.x;       // BB*KG = 20480
  int b = i / KG, k = i % KG;
  float v;
  if (k < 256)      v = xt[b * 256 + k];
  else if (k < 768) v = rdata[b * 512 + (k - 256)];
  else              v = h[b * 512 + (k - 768)];
  Ag[i] = v;
}

// LSTM pointwise; also writes h into A_out[:512]
__global__ void dnc_lstm(const float* __restrict__ gates, float* __restrict__ c,
                         float* __restrict__ h, float* __restrict__ Aout) {
  int i = blockIdx.x * 256 + threadIdx.x;       // BB*HIDN = 8192
  int b = i >> 9, j = i & 511;
  const float* g = gates + (size_t)b * NG;
  float ig = sigm(g[j]);
  float fg = sigm(g[512 + j]);
  float gg = tanhf(g[1024 + j]);
  float og = sigm(g[1536 + j]);
  float cn = fg * c[i] + ig * gg;
  c[i] = cn;
  float hv = og * tanhf(cn);
  h[i] = hv;
  Aout[(size_t)b * KO + j] = hv;
}

// per-batch addressing: psi, usage update, allocation (rank sort + cumprod scan),
// write-key cosine+softmax, write distribution. wdist doubles as prev_w storage.
__global__ void dnc_addr(const float* __restrict__ ctrl, const float* __restrict__ mem,
                         const float* __restrict__ rd, float* __restrict__ usages,
                         float* __restrict__ wdist, float* __restrict__ psi_out)
{
  __shared__ float s_key[WLEN];
  __shared__ float s_buf[NC];
  __shared__ float s_red[NC];
  int b = blockIdx.x, n = threadIdx.x;
  const float* cb = ctrl + (size_t)b * CTRLP;

  float freeg[RR];
  for (int hh = 0; hh < RR; ++hh) freeg[hh] = sigm(cb[3 * WLEN + hh]);      // 384..387
  float wbeta = oneplus_f(cb[3 * WLEN + RR]);                               // 388
  float agate = sigm(cb[3 * WLEN + RR + 1]);                                // 389
  float ggate = sigm(cb[3 * WLEN + RR + 2]);                                // 390
  if (n < WLEN) s_key[n] = cb[n];
  __syncthreads();

  // psi and usage update (prev_w aliased in wdist)
  float psi = 1.0f;
  for (int hh = 0; hh < RR; ++hh)
    psi *= 1.0f - freeg[hh] * rd[((size_t)b * RR + hh) * NC + n];
  psi_out[b * NC + n] = psi;
  float pw = wdist[b * NC + n];
  float u  = usages[b * NC + n];
  u = (u + pw - u * pw) * psi;
  usages[b * NC + n] = u;

  // allocation weights (stable ascending rank sort)
  float u2 = u * (1.0f - EPSF) + EPSF;
  s_buf[n] = u2; __syncthreads();
  int rank = 0;
  for (int m = 0; m < NC; ++m) {
    float um = s_buf[m];
    rank += (um < u2) || (um == u2 && m < n);
  }
  __syncthreads();
  s_buf[rank] = u2;                      // sorted ascending
  __syncthreads();
  float su = s_buf[n];                   // sorted value at position n
  // inclusive cumprod (Hillis-Steele)
  for (int off = 1; off < NC; off <<= 1) {
    float other = (n >= off) ? s_buf[n - off] : 1.0f;
    __syncthreads();
    s_buf[n] *= other;
    __syncthreads();
  }
  float shifted = (n == 0) ? 1.0f : s_buf[n - 1];
  float score = (1.0f - su) * shifted;   // score at sorted position n
  __syncthreads();
  s_red[n] = score; __syncthreads();
  float alloc = s_red[rank];             // scatter back via inverse perm
  __syncthreads();

  // cosine content addressing for write key + softmax
  float kn2 = 0.0f;
  for (int w = 0; w < WLEN; ++w) kn2 += s_key[w] * s_key[w];
  const float* mrow = mem + ((size_t)b * NC + n) * WLEN;
  float dot = 0.0f, mn2 = 0.0f;
  for (int w = 0; w < WLEN; ++w) { float mv = mrow[w]; dot += s_key[w] * mv; mn2 += mv * mv; }
  float sim = dot / (sqrtf(kn2) * sqrtf(mn2) + EPSF) * wbeta;
  float mx = bmax256(sim, s_red);
  float e  = expf(sim - mx);
  float ss = bsum256(e, s_red);
  float cw = e / ss;

  wdist[b * NC + n] = ggate * (agate * alloc + (1.0f - agate) * cw);
}

// mem = mem*psi*(1 - w*erase) + w*write_vec
__global__ void dnc_memupd(const float* __restrict__ ctrl, const float* __restrict__ wdist,
                           const float* __restrict__ psi, float* __restrict__ mem)
{
  int i = blockIdx.x * 256 + threadIdx.x;       // BB*NC*WLEN = 524288
  int b = i >> 15, rem = i & 32767, n = rem >> 7, w = rem & 127;
  const float* cb = ctrl + (size_t)b * CTRLP;
  float er = sigm(cb[WLEN + w]);
  float wv = cb[2 * WLEN + w];
  float wd = wdist[b * NC + n];
  float ps = psi[b * NC + n];
  mem[i] = mem[i] * ps * (1.0f - wd * er) + wd * wv;
}

// link[b,i,j] = ((1-wi-wj)*L + wi*prec_old[j]) * (1 - delta_ij)
__global__ void dnc_linkupd(const float* __restrict__ wdist, const float* __restrict__ prec,
                            float* __restrict__ link)
{
  int idx = blockIdx.x * 256 + threadIdx.x;     // BB*NC*NC = 1048576
  int b = idx >> 16, rem = idx & 65535, i = rem >> 8, j = rem & 255;
  float wi = wdist[b * NC + i], wj = wdist[b * NC + j];
  float L = link[idx];
  L = (1.0f - wi - wj) * L + wi * prec[b * NC + j];
  if (i == j) L = 0.0f;
  link[idx] = L;
}

// prec update + sharpen fwd/bwd in place (per-batch workgroup)
__global__ void dnc_sharpen(const float* __restrict__ ctrl, const float* __restrict__ wdist,
                            float* __restrict__ prec, float* __restrict__ fwd,
                            float* __restrict__ bwd)
{
  __shared__ float s_red[NC];
  int b = blockIdx.x, n = threadIdx.x;

  float wd = wdist[b * NC + n];
  float wsum = bsum256(wd, s_red);
  prec[b * NC + n] = (1.0f - wsum) * prec[b * NC + n] + wd;

  for (int hh = 0; hh < RR; ++hh) {
    size_t idx = ((size_t)b * RR + hh) * NC + n;
    float fac = oneplus_f(ctrl[(size_t)b * CTRLP + 919 + hh]);
    float d = fwd[idx] + EPSF;
    float mx = bmax256(d, s_red);
    d = expf(fac * logf(d / mx));
    float sm = bsum256(d, s_red);
    fwd[idx] = d / sm;

    fac = oneplus_f(ctrl[(size_t)b * CTRLP + 919 + RR + hh]);
    d = bwd[idx] + EPSF;
    mx = bmax256(d, s_red);
    d = expf(fac * logf(d / mx));
    sm = bsum256(d, s_red);
    bwd[idx] = d / sm;
  }
}

// read heads: cosine+softmax, mode mix -> r_dist (becomes prev_rd)
__global__ void dnc_read(const float* __restrict__ ctrl, const float* __restrict__ mem,
                         const float* __restrict__ fwd, const float* __restrict__ bwd,
                         float* __restrict__ rd)
{
  __shared__ float s_key[RR * WLEN];
  __shared__ float s_red[NC];
  __shared__ float s_beta[RR];
  __shared__ float s_m[RR][3];
  int b = blockIdx.x, n = threadIdx.x;
  const float* cb = ctrl + (size_t)b * CTRLP;

  for (int i = n; i < RR * WLEN; i += 256) {
    int hh = i / WLEN, w = i % WLEN;
    s_key[i] = cb[391 + hh * (WLEN + 4) + w];
  }
  if (n < RR) {
    const float* rc = cb + 391 + n * (WLEN + 4);
    s_beta[n] = oneplus_f(rc[WLEN]);
    float m0 = rc[WLEN + 1], m1 = rc[WLEN + 2], m2 = rc[WLEN + 3];
    float mx = fmaxf(m0, fmaxf(m1, m2));
    float e0 = expf(m0 - mx), e1 = expf(m1 - mx), e2 = expf(m2 - mx);
    float s = e0 + e1 + e2;
    s_m[n][0] = e0 / s; s_m[n][1] = e1 / s; s_m[n][2] = e2 / s;
  }
  __syncthreads();

  const float* mrow = mem + ((size_t)b * NC + n) * WLEN;
  float mn2 = 0.0f, dot[RR] = {0, 0, 0, 0};
  for (int w = 0; w < WLEN; ++w) {
    float mv = mrow[w];
    mn2 += mv * mv;
    for (int hh = 0; hh < RR; ++hh) dot[hh] += s_key[hh * WLEN + w] * mv;
  }
  float mnorm = sqrtf(mn2);

  for (int hh = 0; hh < RR; ++hh) {
    float kn2 = 0.0f;
    for (int w = 0; w < WLEN; ++w) { float kv = s_key[hh * WLEN + w]; kn2 += kv * kv; }
    float sim = dot[hh] / (sqrtf(kn2) * mnorm + EPSF) * s_beta[hh];
    float mx = bmax256(sim, s_red);
    float e  = expf(sim - mx);
    float ss = bsum256(e, s_red);
    float cr = e / ss;
    float rv = s_m[hh][0] * bwd[((size_t)b * RR + hh) * NC + n] +
               s_m[hh][1] * cr +
               s_m[hh][2] * fwd[((size_t)b * RR + hh) * NC + n];
    rd[((size_t)b * RR + hh) * NC + n] = rv;      // becomes prev_rd
  }
}

// ---------------- workspace layout (floats) ----------------
static constexpr size_t OFF_W1   = 0;
static constexpr size_t SZ_W1    = (size_t)KG * NG;            // 2,621,440
static constexpr size_t OFF_W2   = OFF_W1 + SZ_W1;
static constexpr size_t SZ_W2    = (size_t)HIDN * CTRLP;       //   475,136
static constexpr size_t OFF_BC   = OFF_W2 + SZ_W2;
static constexpr size_t OFF_W3   = OFF_BC + CTRLP;
static constexpr size_t SZ_W3    = (size_t)KO * OUTS;          //   262,144
static constexpr size_t OFF_BO   = OFF_W3 + SZ_W3;
static constexpr size_t OFF_MEM  = OFF_BO + OUTS;
static constexpr size_t OFF_LINK = OFF_MEM + (size_t)BB * NC * WLEN;
static constexpr size_t OFF_USE  = OFF_LINK + (size_t)BB * NC * NC;
static constexpr size_t OFF_PREC = OFF_USE + BB * NC;
static constexpr size_t OFF_WD   = OFF_PREC + BB * NC;         // w_dist == prev_w
static constexpr size_t OFF_RD   = OFF_WD + BB * NC;           // r_dist == prev_rd
static constexpr size_t OFF_RDAT = OFF_RD + (size_t)BB * RR * NC;
static constexpr size_t OFF_H    = OFF_RDAT + (size_t)BB * RR * WLEN;
static constexpr size_t OFF_C    = OFF_H + BB * HIDN;
static constexpr size_t OFF_PSI  = OFF_C + BB * HIDN;
static constexpr size_t OFF_FWD  = OFF_PSI + BB * NC;
static constexpr size_t OFF_BWD  = OFF_FWD + (size_t)BB * RR * NC;
static constexpr size_t OFF_AG   = OFF_BWD + (size_t)BB * RR * NC;
static constexpr size_t OFF_GATE = OFF_AG + (size_t)BB * KG;
static constexpr size_t OFF_CTRL = OFF_GATE + (size_t)BB * NG;
static constexpr size_t OFF_AO   = OFF_CTRL + (size_t)BB * CTRLP;
static constexpr size_t WS_TOTAL = OFF_AO + (size_t)BB * KO;   // ~20.4 MB

extern "C" void kernel_launch(void* const* d_in, const int* in_sizes, int n_in,
                              void* d_out, int out_size, void* d_ws, size_t ws_size,
                              hipStream_t stream) {
  (void)in_sizes; (void)n_in; (void)out_size; (void)ws_size;
  const float* in_data = (const float*)d_in[0];
  const float* Wx      = (const float*)d_in[1];
  const float* Wh      = (const float*)d_in[2];
  const float* b_lstm  = (const float*)d_in[3];
  const float* Wc      = (const float*)d_in[4];
  const float* bc      = (const float*)d_in[5];
  const float* Wo      = (const float*)d_in[6];
  const float* bo      = (const float*)d_in[7];
  const float* Wr      = (const float*)d_in[8];
  const float* br      = (const float*)d_in[9];
  float* ws  = (float*)d_ws;
  float* out = (float*)d_out;

  // ---- pre-pack fused weights + biases, zero all state/scratch ----
  dnc_pack_w1<<<(int)((SZ_W1 + 255) / 256), 256, 0, stream>>>(Wx, Wh, ws + OFF_W1);
  dnc_pack_w2<<<(int)((SZ_W2 + 255) / 256), 256, 0, stream>>>(Wc, ws + OFF_W2);
  dnc_pack_bc<<<4, 256, 0, stream>>>(bc, ws + OFF_BC);
  dnc_pack_w3<<<(int)((SZ_W3 + 255) / 256), 256, 0, stream>>>(Wo, Wr, ws + OFF_W3);
  dnc_pack_bo<<<1, 256, 0, stream>>>(bo, br, ws + OFF_BO);
  {
    int nz = (int)(WS_TOTAL - OFF_MEM);
    dnc_zero<<<(nz + 255) / 256, 256, 0, stream>>>(ws + OFF_MEM, nz);
  }

  // ---- sequential scan over timesteps ----
  for (int t = 0; t < TST; ++t) {
    const float* xt = in_data + (size_t)t * BB * INS;
    dnc_packA<<<(BB * KG) / 256, 256, 0, stream>>>(xt, ws + OFF_RDAT, ws + OFF_H,
                                                   ws + OFF_AG);
    // gates = [ctrl_in | h] @ [Wx;Wh] + b_lstm    (N=2048 -> 128 tiles, 4 waves/blk)
    dnc_gemm<<<32, 128, 0, stream>>>(ws + OFF_AG, ws + OFF_W1, b_lstm, ws + OFF_GATE,
                                     KG, NG, 0);
    dnc_lstm<<<(BB * HIDN) / 256, 256, 0, stream>>>(ws + OFF_GATE, ws + OFF_C,
                                                    ws + OFF_H, ws + OFF_AO);
    // controls = clip(h @ Wc + bc)                 (N=928 -> 58 tiles)
    dnc_gemm<<<15, 128, 0, stream>>>(ws + OFF_H, ws + OFF_W2, ws + OFF_BC, ws + OFF_CTRL,
                                     HIDN, CTRLP, 1);
    dnc_addr<<<BB, 256, 0, stream>>>(ws + OFF_CTRL, ws + OFF_MEM, ws + OFF_RD,
                                     ws + OFF_USE, ws + OFF_WD, ws + OFF_PSI);
    dnc_memupd<<<(BB * NC * WLEN) / 256, 256, 0, stream>>>(ws + OFF_CTRL, ws + OFF_WD,
                                                           ws + OFF_PSI, ws + OFF_MEM);
    dnc_linkupd<<<(BB * NC * NC) / 256, 256, 0, stream>>>(ws + OFF_WD, ws + OFF_PREC,
                                                          ws + OFF_LINK);
    // fwd/bwd link matvecs on the matrix pipe (uses OLD rd, NEW link)
    dnc_fb_gemm<<<128, 128, 0, stream>>>(ws + OFF_RD, ws + OFF_LINK,
                                         ws + OFF_FWD, ws + OFF_BWD);
    dnc_sharpen<<<BB, 256, 0, stream>>>(ws + OFF_CTRL, ws + OFF_WD, ws + OFF_PREC,
                                        ws + OFF_FWD, ws + OFF_BWD);
    dnc_read<<<BB, 256, 0, stream>>>(ws + OFF_CTRL, ws + OFF_MEM, ws + OFF_FWD,
                                     ws + OFF_BWD, ws + OFF_RD);
    // r_data = r_dist @ mem on the matrix pipe; packs A_out[512:]
    dnc_rdata_gemm<<<32, 128, 0, stream>>>(ws + OFF_RD, ws + OFF_MEM,
                                           ws + OFF_RDAT, ws + OFF_AO);
    // out_t = [h | r_data] @ [Wo;Wr] + (bo+br)     (N=256 -> 16 tiles)
    dnc_gemm<<<4, 128, 0, stream>>>(ws + OFF_AO, ws + OFF_W3, ws + OFF_BO,
                                    out + (size_t)t * BB * OUTS, KO, OUTS, 0);
  }
}